// NTM_read_head_85074712199498
// MI455X (gfx1250) — compile-verified
//
#include <hip/hip_runtime.h>
#include <math.h>

// Problem constants (from reference): B=64, N=32768, M=128, C=1024, out_dim=134
#define BQ   64
#define NQ   32768
#define MQ   128
#define CQ   1024
#define ODIM 134
#define OPAD 144        // 9 tiles of 16
#define EPSQ 1e-16f
#define CHUNK 512       // rows per content-partial block

typedef __attribute__((ext_vector_type(2))) float v2f;
typedef __attribute__((ext_vector_type(8))) float v8f;

__device__ __forceinline__ float softplusf(float x) {
  return x > 20.f ? x : log1pf(__expf(x));
}

// ---------------------------------------------------------------------------
// K1: out[b][o] = emb[b,:] . W_proj[o,:] + b_proj[o]     (64 x 134, K=1024)
// fp32 WMMA 16x16x4. One wave per 16x16 tile. A = emb tile, B = W_proj^T tile.
// A-layout (ISA 7.12.2, 32-bit A 16x4): lane L holds row (L&15), K = 2*(L>>4)+{0,1}
// B-layout mirrors: lane L holds col (L&15), K = 2*(L>>4)+{0,1}; B[k][n]=W_proj[n][k]
// => both are contiguous float2 loads.
// ---------------------------------------------------------------------------
__global__ __launch_bounds__(32) void k_proj(const float* __restrict__ emb,
                                             const float* __restrict__ Wp,
                                             const float* __restrict__ bp,
                                             float* __restrict__ out) {
  const int lane = threadIdx.x;
  const int half = lane >> 4;
  const int lrow = lane & 15;
  const int n0 = blockIdx.x * 16;   // output-feature tile
  const int b0 = blockIdx.y * 16;   // batch tile
  const int ncol = n0 + lrow;

  const float* arow = emb + (size_t)(b0 + lrow) * CQ + 2 * half;
  // clamp OOB feature rows to row 0; garbage columns are masked at the store
  const float* brow = Wp + (size_t)(ncol < ODIM ? ncol : 0) * CQ + 2 * half;

  v8f c = {};
  for (int kk = 0; kk < CQ; kk += 4) {
    v2f a = *(const v2f*)(arow + kk);
    v2f b = *(const v2f*)(brow + kk);
    c = __builtin_amdgcn_wmma_f32_16x16x4_f32(false, a, false, b,
                                              (short)0, c, false, false);
  }
  // D layout: c[r] = D[r + 8*half][lrow]
  if (ncol < ODIM) {
    float bias = bp[ncol];
#pragma unroll
    for (int r = 0; r < 8; ++r) {
      int row = b0 + r + 8 * half;
      out[row * OPAD + ncol] = c[r] + bias;
    }
  }
}

// ---------------------------------------------------------------------------
// K2: per-batch scalar params + k vector + ||k||
// params[b][8] = {beta, g, s0, s1, s2, gamma, knorm, pad}
// ---------------------------------------------------------------------------
__global__ __launch_bounds__(128) void k_params(const float* __restrict__ out,
                                                float* __restrict__ kbuf,
                                                float* __restrict__ params) {
  __shared__ float red[128];
  const int b = blockIdx.x, t = threadIdx.x;
  float kv = out[b * OPAD + t];
  kbuf[b * MQ + t] = kv;
  red[t] = kv * kv;
  __syncthreads();
  for (int s = 64; s > 0; s >>= 1) {
    if (t < s) red[t] += red[t + s];
    __syncthreads();
  }
  if (t == 0) {
    float knorm = sqrtf(red[0]);
    const float* o = out + b * OPAD + MQ;
    float beta = softplusf(o[0]);
    float g = 1.f / (1.f + __expf(-o[1]));
    float a0 = o[2], a1 = o[3], a2 = o[4];
    float mx = fmaxf(a0, fmaxf(a1, a2));
    float e0 = __expf(a0 - mx), e1 = __expf(a1 - mx), e2 = __expf(a2 - mx);
    float inv = 1.f / (e0 + e1 + e2);
    float gamma = 1.f + softplusf(o[5]);
    float* p = params + b * 8;
    p[0] = beta; p[1] = g; p[2] = e0 * inv; p[3] = e1 * inv; p[4] = e2 * inv;
    p[5] = gamma; p[6] = knorm; p[7] = 0.f;
  }
}

// ---------------------------------------------------------------------------
// K3: sim[b][n] = (mem[b,n,:] . k[b,:]) / (||mem[b,n,:]|| * ||k|| + eps)
// Heavy streaming pass over mem (1.07 GB). One wave per 16-row tile.
// WMMA: A = mem tile (16x4 per step), B[k][j] = kvec[k] for all j =>
// every D column holds the dot products; row norm^2 accumulated on VALU from
// the same loaded fragments (lane L covers row L&15, cols == {2h,2h+1} mod 4),
// folded with a single cross-half shuffle.
// ---------------------------------------------------------------------------
__global__ __launch_bounds__(128) void k_sim(const float* __restrict__ mem,
                                             const float* __restrict__ kbuf,
                                             const float* __restrict__ params,
                                             float* __restrict__ sim) {
  const int lane = threadIdx.x & 31;
  const int wave = threadIdx.x >> 5;
  const int half = lane >> 4;
  const int lrow = lane & 15;
  const int b = blockIdx.y;
  const int tile = blockIdx.x * 4 + wave;
  const int n0 = tile * 16;

  const float* arow = mem + ((size_t)b * NQ + n0 + lrow) * MQ + 2 * half;
  const float* krow = kbuf + b * MQ + 2 * half;

  v8f c = {};
  float nrm = 0.f;
  for (int kk = 0; kk < MQ; kk += 4) {
    v2f a  = *(const v2f*)(arow + kk);
    v2f kv = *(const v2f*)(krow + kk);   // wave-uniform per half; L0-resident
    nrm = fmaf(a.x, a.x, nrm);
    nrm = fmaf(a.y, a.y, nrm);
    c = __builtin_amdgcn_wmma_f32_16x16x4_f32(false, a, false, kv,
                                              (short)0, c, false, false);
  }
  nrm += __shfl_xor(nrm, 16, 32);        // lane L now holds ||row (L&15)||^2
  const float knorm = params[b * 8 + 6];
  float* srow = sim + (size_t)b * NQ + n0;
#pragma unroll
  for (int r = 0; r < 8; ++r) {
    int row = r + 8 * half;              // tile-local row carried by c[r]
    float n2 = __shfl(nrm, row, 32);     // convergent
    if (lrow == 0)                       // lanes 0 and 16 each store 8 rows
      srow[row] = c[r] / (sqrtf(n2) * knorm + EPSQ);
  }
}

// ---------------------------------------------------------------------------
// K4: per-batch softmax(beta*sim) -> gate -> circular shift -> pow(gamma) ->
// normalize. One 1024-thread workgroup per b; the full 128 KB w_g row lives in
// LDS (CDNA5: 320 KB per WGP), so neighbor access for the shift is free.
// Deterministic tree reductions (graph replays are bit-stable).
// ---------------------------------------------------------------------------
__global__ __launch_bounds__(1024) void k_weights(const float* __restrict__ sim,
                                                  const float* __restrict__ Wold,
                                                  const float* __restrict__ params,
                                                  float* __restrict__ Wout) {
  extern __shared__ float smem[];        // [NQ] w_g + [1024] reduction scratch
  float* wg  = smem;
  float* red = smem + NQ;
  const int b = blockIdx.x, t = threadIdx.x;
  const float beta = params[b * 8 + 0];
  const float g    = params[b * 8 + 1];
  const float s0   = params[b * 8 + 2];
  const float s1   = params[b * 8 + 3];
  const float s2   = params[b * 8 + 4];
  const float gam  = params[b * 8 + 5];
  const float* srow = sim  + (size_t)b * NQ;
  const float* wold = Wold + (size_t)b * NQ;

  float x[32];
  float lm = -3.4e38f;
#pragma unroll
  for (int i = 0; i < 32; ++i) {
    int n = t + i * 1024;
    x[i] = beta * srow[n];
    lm = fmaxf(lm, x[i]);
  }
  red[t] = lm; __syncthreads();
  for (int s = 512; s > 0; s >>= 1) { if (t < s) red[t] = fmaxf(red[t], red[t + s]); __syncthreads(); }
  float mx = red[0]; __syncthreads();

  float ls = 0.f;
#pragma unroll
  for (int i = 0; i < 32; ++i) { x[i] = __expf(x[i] - mx); ls += x[i]; }
  red[t] = ls; __syncthreads();
  for (int s = 512; s > 0; s >>= 1) { if (t < s) red[t] += red[t + s]; __syncthreads(); }
  float inv = 1.f / red[0]; __syncthreads();

#pragma unroll
  for (int i = 0; i < 32; ++i) {
    int n = t + i * 1024;
    wg[n] = g * x[i] * inv + (1.f - g) * wold[n];
  }
  __syncthreads();

  float lp = 0.f;
#pragma unroll
  for (int i = 0; i < 32; ++i) {
    int n = t + i * 1024;
    float wt = s0 * wg[(n + NQ - 1) & (NQ - 1)] + s1 * wg[n] + s2 * wg[(n + 1) & (NQ - 1)];
    x[i] = powf(wt, gam);
    lp += x[i];
  }
  red[t] = lp; __syncthreads();
  for (int s = 512; s > 0; s >>= 1) { if (t < s) red[t] += red[t + s]; __syncthreads(); }
  float invp = 1.f / (red[0] + EPSQ);

  float* wrow = Wout + (size_t)b * NQ;
#pragma unroll
  for (int i = 0; i < 32; ++i) {
    int n = t + i * 1024;
    wrow[n] = x[i] * invp;
  }
}

// ---------------------------------------------------------------------------
// K5: mem_content partials. Second streaming pass over mem (1.07 GB).
// Thread t == column m; each row is one fully-coalesced 512 B load per block.
// W[n] is wave-uniform (scalar load). global_prefetch for the stream.
// ---------------------------------------------------------------------------
__global__ __launch_bounds__(128) void k_content_partial(const float* __restrict__ mem,
                                                         const float* __restrict__ W,
                                                         float* __restrict__ partial) {
  const int b = blockIdx.y, ch = blockIdx.x, m = threadIdx.x;
  const float* wrow = W + (size_t)b * NQ + (size_t)ch * CHUNK;
  const float* mrow = mem + ((size_t)b * NQ + (size_t)ch * CHUNK) * MQ + m;
  float acc = 0.f;
  for (int j = 0; j < CHUNK; ++j) {
    if ((j & 15) == 0)
      __builtin_prefetch(mrow + (size_t)(j + 32) * MQ, 0, 1); // global_prefetch_b8
    acc = fmaf(wrow[j], mrow[(size_t)j * MQ], acc);
  }
  partial[((size_t)b * (NQ / CHUNK) + ch) * MQ + m] = acc;
}

// K6: deterministic final reduction of the chunk partials.
__global__ __launch_bounds__(128) void k_content_reduce(const float* __restrict__ partial,
                                                        float* __restrict__ outc) {
  const int b = blockIdx.x, m = threadIdx.x;
  float acc = 0.f;
  for (int c = 0; c < NQ / CHUNK; ++c)
    acc += partial[((size_t)b * (NQ / CHUNK) + c) * MQ + m];
  outc[b * MQ + m] = acc;
}

// ---------------------------------------------------------------------------
extern "C" void kernel_launch(void* const* d_in, const int* in_sizes, int n_in,
                              void* d_out, int out_size, void* d_ws, size_t ws_size,
                              hipStream_t stream) {
  const float* emb  = (const float*)d_in[0];  // (64,1024)
  const float* Wold = (const float*)d_in[1];  // (64,32768)
  const float* mem  = (const float*)d_in[2];  // (64,32768,128)
  const float* Wp   = (const float*)d_in[3];  // (134,1024)
  const float* bp   = (const float*)d_in[4];  // (134,)

  float* Wout = (float*)d_out;                // first 64*32768 floats
  float* outc = Wout + (size_t)BQ * NQ;       // then 64*128 floats

  // workspace layout (bytes)
  char* ws = (char*)d_ws;
  float* proj    = (float*)(ws + 0);                    //  64*144*4  = 36864
  float* kbuf    = (float*)(ws + 36864);                //  64*128*4  = 32768
  float* params  = (float*)(ws + 69632);                //  64*8*4    = 2048
  float* sim     = (float*)(ws + 131072);               //  64*32768*4 = 8 MiB
  float* partial = (float*)(ws + 131072 + (size_t)BQ * NQ * 4); // 64*64*128*4 = 2 MiB
  (void)ws_size; (void)in_sizes; (void)n_in; (void)out_size;

  // K1: projection GEMM (WMMA f32 16x16x4)
  k_proj<<<dim3(OPAD / 16, BQ / 16), 32, 0, stream>>>(emb, Wp, bp, proj);

  // K2: per-batch params
  k_params<<<BQ, 128, 0, stream>>>(proj, kbuf, params);

  // K3: cosine-similarity streaming pass (WMMA in the hot loop)
  k_sim<<<dim3(NQ / 16 / 4, BQ), 128, 0, stream>>>(mem, kbuf, params, sim);

  // K4: softmax/gate/shift/sharpen, 128 KB w_g row in LDS
  size_t shmem = (size_t)(NQ + 1024) * sizeof(float);   // 135168 B < 320 KB
  (void)hipFuncSetAttribute((const void*)k_weights,
                            hipFuncAttributeMaxDynamicSharedMemorySize,
                            (int)shmem);
  k_weights<<<BQ, 1024, shmem, stream>>>(sim, Wold, params, Wout);

  // K5+K6: mem_content via deterministic two-stage reduction
  k_content_partial<<<dim3(NQ / CHUNK, BQ), MQ, 0, stream>>>(mem, Wout, partial);
  k_content_reduce<<<BQ, MQ, 0, stream>>>(partial, outc);
}